// OffloadableEmbedding_72155450573263
// MI455X (gfx1250) — compile-verified
//
#include <hip/hip_runtime.h>

// Embedding gather: out[r, 0:128] = weight[idx[r], 0:128]
// Row = 128 fp32 = 512 B = 32 lanes (wave32) x float4 (16 B).
// One wave per row, one global_load_b128 + one NT global_store_b128 per lane.
// Pure bandwidth problem (~840 MB @ 23.3 TB/s ~= 36 us): no WMMA applicable.
// gfx1250-specific paths used: global_prefetch_b8 (next-row prefetch) and
// non-temporal stores (keep the 192MB L2 reserved for the 512MB weight table).

typedef float v4f __attribute__((ext_vector_type(4)));

__global__ __launch_bounds__(256) void emb_gather_kernel(
    const int* __restrict__ indices,   // [nrows]
    const v4f* __restrict__ weight,    // [num_emb * 32] float4 (row = 32 x float4)
    v4f* __restrict__ out,             // [nrows * 32] float4
    int nrows) {
  const int lane  = threadIdx.x & 31;
  const int wave  = (int)((blockIdx.x * blockDim.x + threadIdx.x) >> 5);
  const int nwave = (int)((gridDim.x * blockDim.x) >> 5);

  int r = wave;
  if (r >= nrows) return;

  // Pointer-chase pipeline: index for row r is already resolved when we enter
  // the iteration; while the 512B row load is in flight we resolve the next
  // index and prefetch the next row (global_prefetch_b8, no counter cost).
  int e = indices[r];  // wave-uniform (all 32 lanes load same dword, L0 hit)

  while (true) {
    const v4f* src = &weight[(size_t)e * 32 + lane];
    v4f row = *src;  // global_load_b128, coalesced 512B per wave

    const int rn = r + nwave;
    if (rn < nrows) {
      e = indices[rn];
      // Speculative prefetch of next row's 16B chunk for this lane.
      __builtin_prefetch(&weight[(size_t)e * 32 + lane], 0, 1);
    }

    // Non-temporal store: output is write-once / never re-read here; keep it
    // out of L2 so L2 capacity caches the weight table (random-gather reuse).
    __builtin_nontemporal_store(row, &out[(size_t)r * 32 + lane]);

    if (rn >= nrows) break;
    r = rn;
  }
}

extern "C" void kernel_launch(void* const* d_in, const int* in_sizes, int n_in,
                              void* d_out, int out_size, void* d_ws, size_t ws_size,
                              hipStream_t stream) {
  const int* indices = (const int*)d_in[0];   // int32 indices, flat [4096*200]
  const v4f* weight  = (const v4f*)d_in[1];   // fp32 weight [1e6 x 128] as float4
  v4f*       out     = (v4f*)d_out;           // fp32 out [nrows x 128] as float4

  const int nrows = in_sizes[0];              // 819200

  // 4096 blocks x 256 threads = 32768 waves (8 waves per 256-thread block,
  // wave32). Grid-stride: ~25 rows per wave -> prefetch pipeline has depth.
  const int threads = 256;
  const int blocks  = 4096;
  emb_gather_kernel<<<blocks, threads, 0, stream>>>(indices, weight, out, nrows);
}